// density_computer_4647154614875
// MI455X (gfx1250) — compile-verified
//
#include <hip/hip_runtime.h>
#include <hip/hip_bf16.h>
#include <stdint.h>

// ---------------------------------------------------------------------------
// Gaussian-KDE density histogram (MI455X / gfx1250, wave32)
//   hist[j] = sum over all (t,n) with z_mask[n] of exp(-0.5*((q - r_j)/dr)^2)
//   out = [ r_list (598) , hist/hist.sum() (598) ]
// norm/dr/T factors cancel in the normalization, so they are omitted.
// f32 exp underflows to 0 for |q - r| > ~1.44 => +/-16 bin window is exact.
// ---------------------------------------------------------------------------

#define R_BINS   598
#define R_PAD    640      // padded to a multiple of 64 for the WMMA reduction
#define HALF_WIN 16

typedef __attribute__((ext_vector_type(2))) float v2f;
typedef __attribute__((ext_vector_type(8))) float v8f;

__global__ void zero_acc_kernel(float* __restrict__ acc) {
    int i = blockIdx.x * blockDim.x + threadIdx.x;
    if (i < R_PAD) acc[i] = 0.0f;
}

__global__ __launch_bounds__(256)
void density_accum_kernel(const float* __restrict__ traj,
                          const uint8_t* __restrict__ zmask,
                          float* __restrict__ acc,
                          int total, int nPart) {
    __shared__ float hist[R_PAD];
    for (int j = threadIdx.x; j < R_PAD; j += blockDim.x) hist[j] = 0.0f;
    __syncthreads();

    const int stride = gridDim.x * blockDim.x;
    for (int i = blockIdx.x * blockDim.x + threadIdx.x; i < total; i += stride) {
        const int n = i % nPart;
        if (zmask[n]) {
            const float q = traj[3 * i + 2];            // AXIS = 2
            const float u = (q - 0.05f) * 10.0f;        // (q - r_0) / dr
            const int jc  = (int)floorf(u + 0.5f);
            int jlo = jc - HALF_WIN; if (jlo < 0) jlo = 0;
            int jhi = jc + HALF_WIN; if (jhi > R_BINS - 1) jhi = R_BINS - 1;
            #pragma unroll 4
            for (int j = jlo; j <= jhi; ++j) {
                const float d = u - (float)j;           // (q - r_j)/dr
                const float g = __expf(-0.5f * d * d);  // v_exp_f32
                atomicAdd(&hist[j], g);                 // ds_add_f32
            }
        }
    }
    __syncthreads();

    // one flush per block: 768 blocks x 598 bins of global f32 atomics
    for (int j = threadIdx.x; j < R_BINS; j += blockDim.x) {
        const float h = hist[j];
        if (h != 0.0f) atomicAdd(&acc[j], h);
    }
}

// Single wave32, EXEC all ones (required by WMMA).
// Grand total via V_WMMA_F32_16X16X4_F32: A = 16x4 ones, B = 64 data values
// per issue, C carries the f32 accumulator. All D rows are identical column
// sums, so reducing c[0] across 32 lanes gives 2x the total (rows 0 and 8).
__global__ void finalize_kernel(const float* __restrict__ acc,
                                float* __restrict__ out) {
    const int lane = threadIdx.x;   // 0..31

    v8f c = {};
    v2f a; a.x = 1.0f; a.y = 1.0f;  // all-ones A matrix (2 VGPRs / lane)

    #pragma unroll
    for (int t = 0; t < R_PAD / 64; ++t) {
        v2f b;
        b.x = acc[t * 64 + lane];
        b.y = acc[t * 64 + 32 + lane];
        // (neg_a, A, neg_b, B, c_mod, C, reuse_a, reuse_b)
        c = __builtin_amdgcn_wmma_f32_16x16x4_f32(
                false, a, false, b, (short)0, c, false, false);
    }

    float s = c[0];
    #pragma unroll
    for (int off = 16; off >= 1; off >>= 1)
        s += __shfl_xor(s, off, 32);
    const float inv_total = 2.0f / s;   // s == 2 * sum(acc)

    for (int j = lane; j < R_BINS; j += 32) {
        out[j]          = 0.05f + 0.1f * (float)j;   // r_list
        out[R_BINS + j] = acc[j] * inv_total;        // P / P.sum()
    }
}

extern "C" void kernel_launch(void* const* d_in, const int* in_sizes, int n_in,
                              void* d_out, int out_size, void* d_ws, size_t ws_size,
                              hipStream_t stream) {
    const float*   traj  = (const float*)d_in[0];     // (T, N, 3) f32
    const uint8_t* zmask = (const uint8_t*)d_in[1];   // (N,) bool (1 byte)
    float*         out   = (float*)d_out;             // 1196 f32
    float*         acc   = (float*)d_ws;              // 640 f32 scratch

    const int nPart = in_sizes[1];        // N
    const int total = in_sizes[0] / 3;    // T*N particles

    zero_acc_kernel<<<(R_PAD + 255) / 256, 256, 0, stream>>>(acc);
    density_accum_kernel<<<768, 256, 0, stream>>>(traj, zmask, acc, total, nPart);
    finalize_kernel<<<1, 32, 0, stream>>>(acc, out);
}